// Model_4303557231029
// MI455X (gfx1250) — compile-verified
//
#include <hip/hip_runtime.h>
#include <hip/hip_bf16.h>
#include <stdint.h>

#define B_      64
#define C_      3
#define H_      128
#define W_      416
#define HW_     (H_*W_)
#define CHW_    (C_*HW_)
#define TOPK_   100
#define BH_     16
#define BANDS_  (H_/BH_)
#define NBINS_  4096
#define CANDMAX_ 1024

#if defined(__has_builtin)
# if __has_builtin(__builtin_amdgcn_tensor_load_to_lds)
#  define HAS_TDM 1
# endif
# if __has_builtin(__builtin_amdgcn_wmma_f32_16x16x4_f32)
#  define HAS_WMMA 1
# endif
#endif
#ifndef HAS_TDM
# define HAS_TDM 0
#endif
#ifndef HAS_WMMA
# define HAS_WMMA 0
#endif

typedef unsigned int v4u_t __attribute__((ext_vector_type(4)));
typedef int          v8i_t __attribute__((ext_vector_type(8)));
typedef int          v4i_t __attribute__((ext_vector_type(4)));
typedef float        v2f_t __attribute__((ext_vector_type(2)));
typedef float        v8f_t __attribute__((ext_vector_type(8)));

__device__ __constant__ float c_dimref[9] = {
    3.88f, 1.63f, 1.53f,
    0.84f, 1.76f, 0.66f,
    1.78f, 1.70f, 0.58f};

// ---------------------------------------------------------------------------
// TDM: 2-D tile load (nrows x W_ f32, row stride W_ elements) into LDS.
// Descriptor layout per CDNA5 ISA.
// ---------------------------------------------------------------------------
#if HAS_TDM
__device__ __forceinline__ void tdm_load_2d(const float* gsrc, float* ldst, int nrows) {
  uint64_t ga   = (uint64_t)(uintptr_t)gsrc;
  uint32_t loff = (uint32_t)(uintptr_t)ldst;   // addrspace(3) ptr -> LDS byte offset
  v4u_t g0;
  g0[0] = 1u;                                   // count=1, user mode, no gather
  g0[1] = loff;                                 // lds_addr
  g0[2] = (uint32_t)ga;                         // global_addr[31:0]
  g0[3] = (uint32_t)((ga >> 32) & 0x01FFFFFFull) | 0x80000000u; // addr[56:32], type=2
  v8i_t g1;
  g1[0] = (int)(2u << 16);                                      // data_size = 4 bytes
  g1[1] = (int)(((uint32_t)W_ & 0xFFFFu) << 16);                // tensor_dim0[15:0]
  g1[2] = (int)((((uint32_t)W_) >> 16) |
                (((uint32_t)nrows & 0xFFFFu) << 16));           // dim0 hi / dim1 lo
  g1[3] = (int)((((uint32_t)nrows) >> 16) |
                (((uint32_t)W_) << 16));                        // dim1 hi / tile_dim0
  g1[4] = (int)((uint32_t)nrows & 0xFFFFu);                     // tile_dim1 (tile_dim2=0)
  g1[5] = (int)W_;                                              // tensor_dim0_stride lo32
  g1[6] = 0;
  g1[7] = 0;
  v4i_t z4 = {};
#if __clang_major__ >= 23
  v8i_t z8 = {};
  __builtin_amdgcn_tensor_load_to_lds(g0, g1, z4, z4, z8, 0);
#else
  __builtin_amdgcn_tensor_load_to_lds(g0, g1, z4, z4, 0);
#endif
}
#endif

// ---------------------------------------------------------------------------
__global__ void k_zero(unsigned* __restrict__ p, int n) {
  int i = blockIdx.x * 256 + threadIdx.x;
  if (i < n) p[i] = 0u;
}

// ---------------------------------------------------------------------------
// mode 0: histogram peak scores.  mode 1: compact peaks with bin >= cutoff[b].
// One block per (band, channel, image). Heatmap tile streamed via TDM to LDS.
// ---------------------------------------------------------------------------
__global__ __launch_bounds__(256) void k_band(const float* __restrict__ kf,
                                              unsigned* __restrict__ hist,
                                              const unsigned* __restrict__ cutoff,
                                              unsigned* __restrict__ ccnt,
                                              unsigned* __restrict__ cand,
                                              int mode) {
  __shared__ float tile[(BH_ + 2) * W_];
  const int band = blockIdx.x, c = blockIdx.y, b = blockIdx.z;
  const int tid = threadIdx.x;
  const int r0 = band * BH_;
  const int row_start = (r0 == 0) ? 0 : (r0 - 1);
  const int row_end   = (r0 + BH_ + 1 > H_) ? H_ : (r0 + BH_ + 1);
  const int nrows     = row_end - row_start;
  const float* gsrc = kf + ((size_t)(b * C_ + c) * H_ + row_start) * W_;
  float* ldst = tile + (row_start - (r0 - 1)) * W_;

#if HAS_TDM
  if (tid < 32) {                       // wave 0 issues the DMA; TENSORcnt is per-wave
    tdm_load_2d(gsrc, ldst, nrows);
    __builtin_amdgcn_s_wait_tensorcnt(0);
  }
#else
  for (int i = tid; i < nrows * W_; i += 256) ldst[i] = gsrc[i];
#endif
  if (r0 == 0)
    for (int x = tid; x < W_; x += 256) tile[x] = -__builtin_inff();
  if (r0 + BH_ == H_)
    for (int x = tid; x < W_; x += 256) tile[(BH_ + 1) * W_ + x] = -__builtin_inff();
  __syncthreads();

  unsigned cb = 0;
  if (mode) cb = cutoff[b];

  auto at = [&](int rr, int xx) -> float {
    return (xx < 0 || xx >= W_) ? -__builtin_inff() : tile[rr * W_ + xx];
  };

  for (int i = tid; i < BH_ * W_; i += 256) {
    int ry = i / W_;
    int x  = i - ry * W_;
    int r  = ry + 1;
    float v  = tile[r * W_ + x];
    float nm = at(r - 1, x - 1);
    nm = fmaxf(nm, at(r - 1, x));
    nm = fmaxf(nm, at(r - 1, x + 1));
    nm = fmaxf(nm, at(r,     x - 1));
    nm = fmaxf(nm, at(r,     x + 1));
    nm = fmaxf(nm, at(r + 1, x - 1));
    nm = fmaxf(nm, at(r + 1, x));
    nm = fmaxf(nm, at(r + 1, x + 1));
    if (v >= nm) {                       // NMS peak (sigmoid is monotone in logit)
      float s = 1.0f / (1.0f + expf(-v));
      unsigned bin = (unsigned)(s * (float)NBINS_);
      if (bin > NBINS_ - 1) bin = NBINS_ - 1;
      if (mode == 0) {
        atomicAdd(&hist[(size_t)b * NBINS_ + bin], 1u);
      } else if (bin >= cb) {
        unsigned pos = atomicAdd(&ccnt[b], 1u);
        if (pos < CANDMAX_) {
          unsigned idx = (unsigned)(c * HW_ + (r0 + ry) * W_ + x);
          cand[((size_t)b * CANDMAX_ + pos) * 2 + 0] = __float_as_uint(s);
          cand[((size_t)b * CANDMAX_ + pos) * 2 + 1] = idx;
        }
      }
    }
  }
}

// ---------------------------------------------------------------------------
__global__ void k_cutoff(const unsigned* __restrict__ hist, unsigned* __restrict__ cutoff) {
  int b = threadIdx.x;
  if (b >= B_) return;
  unsigned cum = 0;
  int binres = 0;
  for (int bin = NBINS_ - 1; bin >= 0; --bin) {
    cum += hist[(size_t)b * NBINS_ + bin];
    if (cum >= TOPK_) { binres = bin; break; }
  }
  cutoff[b] = (unsigned)binres;
}

// ---------------------------------------------------------------------------
// Per-image bitonic sort (descending) of up to 1024 candidates; tie-break on
// lower index via key = (scoreBits<<32) | ~idx.  Top-100 written to ws.
// ---------------------------------------------------------------------------
__global__ __launch_bounds__(CANDMAX_) void k_sort(const unsigned* __restrict__ cand,
                                                   const unsigned* __restrict__ ccnt,
                                                   unsigned* __restrict__ top) {
  __shared__ unsigned long long keys[CANDMAX_];
  const int b = blockIdx.x, tid = threadIdx.x;
  unsigned n = ccnt[b];
  if (n > CANDMAX_) n = CANDMAX_;
  unsigned long long key = 0ull;
  if (tid < (int)n) {
    unsigned s  = cand[((size_t)b * CANDMAX_ + tid) * 2 + 0];
    unsigned ix = cand[((size_t)b * CANDMAX_ + tid) * 2 + 1];
    key = ((unsigned long long)s << 32) | (unsigned long long)(0xFFFFFFFFu - ix);
  }
  keys[tid] = key;
  for (int k = 2; k <= CANDMAX_; k <<= 1) {
    for (int j = k >> 1; j > 0; j >>= 1) {
      __syncthreads();
      int ixj = tid ^ j;
      if (ixj > tid) {
        unsigned long long a = keys[tid], c2 = keys[ixj];
        bool desc = ((tid & k) == 0);
        if ((a < c2) == desc) { keys[tid] = c2; keys[ixj] = a; }
      }
    }
  }
  __syncthreads();
  if (tid < TOPK_) {
    unsigned long long kk = keys[tid];
    top[((size_t)b * TOPK_ + tid) * 2 + 0] = (unsigned)(kk >> 32);
    top[((size_t)b * TOPK_ + tid) * 2 + 1] = 0xFFFFFFFFu - (unsigned)(kk & 0xFFFFFFFFu);
  }
}

// ---------------------------------------------------------------------------
// Decode: 1 wave per image, 16 detections per WMMA tile.
// locs ray = uv1 @ Kinv^T via V_WMMA_F32_16X16X4_F32 (A:16x4, B:4x16, D:16x16).
// ---------------------------------------------------------------------------
__global__ __launch_bounds__(32) void k_decode(const float* __restrict__ off_fr,
                                               const float* __restrict__ main_off,
                                               const float* __restrict__ pred3d,
                                               const float* __restrict__ Ks,
                                               const unsigned* __restrict__ top,
                                               float* __restrict__ res,
                                               float* __restrict__ valid) {
  const int b = blockIdx.x, lane = threadIdx.x;
  const float* K = Ks + (size_t)b * 9;
  float a00=K[0],a01=K[1],a02=K[2],a10=K[3],a11=K[4],a12=K[5],a20=K[6],a21=K[7],a22=K[8];
  float det = a00*(a11*a22 - a12*a21) - a01*(a10*a22 - a12*a20) + a02*(a10*a21 - a11*a20);
  float rd = 1.0f / det;
  float inv[3][3];
  inv[0][0]=(a11*a22-a12*a21)*rd; inv[0][1]=(a02*a21-a01*a22)*rd; inv[0][2]=(a01*a12-a02*a11)*rd;
  inv[1][0]=(a12*a20-a10*a22)*rd; inv[1][1]=(a00*a22-a02*a20)*rd; inv[1][2]=(a02*a10-a00*a12)*rd;
  inv[2][0]=(a10*a21-a11*a20)*rd; inv[2][1]=(a01*a20-a00*a21)*rd; inv[2][2]=(a00*a11-a01*a10)*rd;

  __shared__ float dtile[256];
  const int m = lane & 15;

  for (int t = 0; t < (TOPK_ + 15) / 16; ++t) {
    const int d = t * 16 + m;
    const bool live = (lane < 16) && (d < TOPK_);
    unsigned sbits = 0, idx = 0;
    if (live) {
      sbits = top[((size_t)b * TOPK_ + d) * 2 + 0];
      idx   = top[((size_t)b * TOPK_ + d) * 2 + 1];
    }
    if (idx >= (unsigned)CHW_) idx = 0;   // padded/invalid slots
    const float score = __uint_as_float(sbits);
    const unsigned cls = idx / HW_;
    const unsigned rem = idx - cls * HW_;
    const unsigned yy  = rem / W_;
    const unsigned xx  = rem - yy * W_;
    const size_t pos = (size_t)yy * W_ + xx;

    const float* mo = main_off + (size_t)b * 2 * HW_ + pos;
    const float mx = (float)xx + 1.0f / (1.0f + expf(-mo[0]));
    const float my = (float)yy + 1.0f / (1.0f + expf(-mo[HW_]));

    // A operand (16x4 f32): lanes 0-15 hold K=0,1 of row m; lanes 16-31 hold K=2,3 (=1,0).
    v2f_t av;
    av.x = (lane < 16) ? mx * 4.0f : 1.0f;
    av.y = (lane < 16) ? my * 4.0f : 0.0f;
    // B operand (4x16 f32): B[k][n] = Kinv[n][k]; lanes 0-15 -> K=0,1; 16-31 -> K=2,3.
    v2f_t bv;
    {
      float b0 = 0.f, b1 = 0.f;
      if (m < 3) {
        if (lane < 16) { b0 = inv[m][0]; b1 = inv[m][1]; }
        else           { b0 = inv[m][2]; b1 = 0.0f;      }
      }
      bv.x = b0; bv.y = b1;
    }
#if HAS_WMMA
    {
      v8f_t cv = {};
      v8f_t dv = __builtin_amdgcn_wmma_f32_16x16x4_f32(
          /*neg_a=*/false, av, /*neg_b=*/false, bv,
          /*c_mod=*/(short)0, cv, /*reuse_a=*/false, /*reuse_b=*/false);
      const int mb  = (lane < 16) ? 0 : 8;
      const int col = m;
      #pragma unroll
      for (int r = 0; r < 8; ++r) dtile[(mb + r) * 16 + col] = dv[r];
    }
#else
    if (lane < 16) {
      const float u0 = mx * 4.0f, u1 = my * 4.0f;
      for (int j = 0; j < 3; ++j)
        dtile[lane * 16 + j] = u0 * inv[j][0] + u1 * inv[j][1] + inv[j][2];
    }
#endif
    __syncthreads();

    if (live) {
      const float* p3 = pred3d + (size_t)b * 6 * HW_ + pos;
      const float p0  = p3[0];
      const float pd1 = p3[(size_t)1 * HW_], pd2 = p3[(size_t)2 * HW_], pd3 = p3[(size_t)3 * HW_];
      const float p4  = p3[(size_t)4 * HW_], p5  = p3[(size_t)5 * HW_];
      const float depth = 28.01f + p0 * 16.32f;
      const float dim0 = c_dimref[cls * 3 + 0] * expf(pd1);
      const float dim1 = c_dimref[cls * 3 + 1] * expf(pd2);
      const float dim2 = c_dimref[cls * 3 + 2] * expf(pd3);
      const float alpha = atan2f(p4, p5);
      const float l0 = depth * dtile[m * 16 + 0];
      const float l1 = depth * dtile[m * 16 + 1];
      const float l2 = depth * dtile[m * 16 + 2];
      const float ryv = alpha + atan2f(l0, l2);

      const float* ofr = off_fr + (size_t)b * 16 * HW_ + pos;
      float vpx[8], vpy[8];
      float mnx = 3.4e38f, mny = 3.4e38f, mxx = -3.4e38f, mxy = -3.4e38f;
      #pragma unroll
      for (int v = 0; v < 8; ++v) {
        const float ox = ofr[(size_t)(2 * v + 0) * HW_];
        const float oy = ofr[(size_t)(2 * v + 1) * HW_];
        const float px = (ox + mx) * 4.0f;
        const float py = (oy + my) * 4.0f;
        vpx[v] = px; vpy[v] = py;
        mnx = fminf(mnx, px); mny = fminf(mny, py);
        mxx = fmaxf(mxx, px); mxy = fmaxf(mxy, py);
      }
      float* row = res + ((size_t)b * TOPK_ + d) * 30;
      row[0]  = (float)cls;
      row[1]  = alpha;
      row[2]  = mnx; row[3] = mny; row[4] = mxx; row[5] = mxy;
      row[6]  = dim0; row[7] = dim1; row[8] = dim2;
      row[9]  = l0; row[10] = l1; row[11] = l2;
      row[12] = ryv;
      row[13] = score;
      #pragma unroll
      for (int v = 0; v < 8; ++v) { row[14 + 2 * v] = vpx[v]; row[15 + 2 * v] = vpy[v]; }
      valid[(size_t)b * TOPK_ + d] = (score > 0.25f) ? 1.0f : 0.0f;
    }
    __syncthreads();
  }
}

// ---------------------------------------------------------------------------
extern "C" void kernel_launch(void* const* d_in, const int* in_sizes, int n_in,
                              void* d_out, int out_size, void* d_ws, size_t ws_size,
                              hipStream_t stream) {
  const float* main_kf  = (const float*)d_in[0];
  const float* off_fr   = (const float*)d_in[1];
  const float* main_off = (const float*)d_in[2];
  const float* pred3d   = (const float*)d_in[3];
  const float* Ks       = (const float*)d_in[4];
  float* res   = (float*)d_out;
  float* valid = res + (size_t)B_ * TOPK_ * 30;

  // Workspace layout (~1.6 MB)
  unsigned* hist   = (unsigned*)d_ws;                    // B_*NBINS_
  unsigned* cutoff = hist + (size_t)B_ * NBINS_;         // B_
  unsigned* ccnt   = cutoff + B_;                        // B_
  unsigned* cand   = ccnt + B_;                          // B_*CANDMAX_*2
  unsigned* top    = cand + (size_t)B_ * CANDMAX_ * 2;   // B_*TOPK_*2

  const int nz = B_ * NBINS_ + 2 * B_;
  k_zero<<<(nz + 255) / 256, 256, 0, stream>>>(hist, nz);

  dim3 gband(BANDS_, C_, B_);
  k_band<<<gband, 256, 0, stream>>>(main_kf, hist, nullptr, nullptr, nullptr, 0);
  k_cutoff<<<1, 64, 0, stream>>>(hist, cutoff);
  k_band<<<gband, 256, 0, stream>>>(main_kf, hist, cutoff, ccnt, cand, 1);
  k_sort<<<B_, CANDMAX_, 0, stream>>>(cand, ccnt, top);
  k_decode<<<B_, 32, 0, stream>>>(off_fr, main_off, pred3d, Ks, top, res, valid);
}